// MyCheb_29386166239372
// MI455X (gfx1250) — compile-verified
//
#include <hip/hip_runtime.h>
#include <stdint.h>

typedef float v2f __attribute__((ext_vector_type(2)));
typedef float v4f __attribute__((ext_vector_type(4)));
typedef float v8f __attribute__((ext_vector_type(8)));

#define P_KEEP     0.4f   // 1 - 0.6
#define DROP_SCALE 2.5f   // 1 / (1 - 0.6)

// ---------------------------------------------------------------- utilities

__device__ __forceinline__ uint32_t hash_u32(uint32_t x, uint32_t seed) {
  uint32_t h = x * 2654435761u + seed * 0x9E3779B9u + 0x7F4A7C15u;
  h ^= h >> 16; h *= 0x85EBCA6Bu;
  h ^= h >> 13; h *= 0xC2B2AE35u;
  h ^= h >> 16;
  return h;
}

__global__ void zero4_f32(float* __restrict__ p, int n4) {
  int i = blockIdx.x * blockDim.x + threadIdx.x;
  if (i < n4) *(v4f*)(p + 4 * (size_t)i) = (v4f){0.f, 0.f, 0.f, 0.f};
}

// ---------------------------------------------------------------- edge weights

__global__ void deg_kernel(const long long* __restrict__ row,
                           float* __restrict__ deg, int E) {
  int e = blockIdx.x * blockDim.x + threadIdx.x;
  if (e < E) unsafeAtomicAdd(&deg[(int)row[e]], 1.0f);
}

__global__ void dis_kernel(float* __restrict__ deg, int n) {
  int i = blockIdx.x * blockDim.x + threadIdx.x;
  if (i < n) {
    float d = deg[i];
    deg[i] = (d > 0.0f) ? rsqrtf(fmaxf(d, 1.0f)) : 0.0f;   // in-place deg -> dis
  }
}

__global__ void wgt_kernel(const long long* __restrict__ row,
                           const long long* __restrict__ col,
                           const float* __restrict__ dis,
                           float* __restrict__ w, int E) {
  int e = blockIdx.x * blockDim.x + threadIdx.x;
  if (e < E) w[e] = -dis[(int)row[e]] * dis[(int)col[e]];
}

// ---------------------------------------------------------------- dropout (vectorized b128)

__global__ void dropout4_kernel(const float* __restrict__ x, float* __restrict__ y,
                                int n4, uint32_t seed) {
  int i = blockIdx.x * blockDim.x + threadIdx.x;
  if (i < n4) {
    v4f xv = *(const v4f*)(x + 4 * (size_t)i);
    v4f yv;
#pragma unroll
    for (int j = 0; j < 4; ++j) {
      uint32_t h = hash_u32((uint32_t)(4 * i + j), seed);
      float r = (float)(h >> 8) * (1.0f / 16777216.0f);
      yv[j] = (r < P_KEEP) ? xv[j] * DROP_SCALE : 0.0f;
    }
    *(v4f*)(y + 4 * (size_t)i) = yv;
  }
}

// ---------------------------------------------------------------- SpMM: tx1[row] += w * xd[col]
// 16 consecutive lanes cover one edge's full 64-float feature row (coalesced
// 256B gather from L2-resident xd), then 4 f32 atomics each.

__global__ void prop_kernel(const float* __restrict__ xd,
                            const long long* __restrict__ row,
                            const long long* __restrict__ col,
                            const float* __restrict__ w,
                            float* __restrict__ tx1, int E) {
  const int GPE = 16;                       // float4 groups per edge (Fin=64)
  int tid = blockIdx.x * blockDim.x + threadIdx.x;
  int e = tid >> 4;
  if (e >= E) return;
  int f = (tid & (GPE - 1)) << 2;
  int r = (int)row[e];
  int c = (int)col[e];
  float we = w[e];
  v4f xv = *(const v4f*)(xd + (size_t)c * 64 + f);
  float* dst = tx1 + (size_t)r * 64 + f;
  unsafeAtomicAdd(dst + 0, we * xv.x);
  unsafeAtomicAdd(dst + 1, we * xv.y);
  unsafeAtomicAdd(dst + 2, we * xv.z);
  unsafeAtomicAdd(dst + 3, we * xv.w);
}

// ---------------------------------------------------------------- fused dual-GEMM via f32 WMMA
// out = A0 @ W0 + A1 @ W1 + b (+ ReLU), one wave per 16x16 output tile.
// Compile-time FIN/FOUT -> fully unrolled straight-line WMMA stream with
// immediate-offset loads/stores.
// A frag (16x4 f32): lane L holds A[L%16][2*(L>>4)+j] in component j (one b64 load)
// B frag (4x16 f32): lane L holds B[2*(L>>4)+j][L%16] in component j
// D (16x16 f32):     vgpr r, lane L = D[r + 8*(L>>4)][L%16]
// Requires N % 16 == 0 (N = 50000 = 3125 * 16).

template <int FIN, int FOUT, int RELU>
__global__ void cheb_gemm_wmma(const float* __restrict__ A0,
                               const float* __restrict__ A1,
                               const float* __restrict__ W0,
                               const float* __restrict__ W1,
                               const float* __restrict__ bias,
                               float* __restrict__ out) {
  int lane = threadIdx.x & 31;
  int wave = threadIdx.x >> 5;
  int row0 = blockIdx.x * 16;
  int col0 = wave * 16;
  int m    = lane & 15;
  int half = lane >> 4;
  int n    = col0 + m;

  const float* arow0 = A0 + (size_t)(row0 + m) * FIN + 2 * half;
  const float* arow1 = A1 + (size_t)(row0 + m) * FIN + 2 * half;
  const float* w0p   = W0 + (size_t)(2 * half) * FOUT + n;
  const float* w1p   = W1 + (size_t)(2 * half) * FOUT + n;

  v8f acc = {};
#pragma unroll
  for (int k = 0; k < FIN; k += 4) {
    v2f a0 = *(const v2f*)(arow0 + k);
    v2f a1 = *(const v2f*)(arow1 + k);
    v2f b0, b1;
    b0.x = w0p[k * FOUT];
    b0.y = w0p[k * FOUT + FOUT];
    b1.x = w1p[k * FOUT];
    b1.y = w1p[k * FOUT + FOUT];
    acc = __builtin_amdgcn_wmma_f32_16x16x4_f32(false, a0, false, b0,
                                                (short)0, acc, false, false);
    acc = __builtin_amdgcn_wmma_f32_16x16x4_f32(false, a1, false, b1,
                                                (short)0, acc, false, false);
  }

  float  bv    = bias[n];
  float* obase = out + (size_t)(row0 + 8 * half) * FOUT + n;
#pragma unroll
  for (int r = 0; r < 8; ++r) {
    float v = acc[r] + bv;
    if (RELU) v = fmaxf(v, 0.0f);
    obase[r * FOUT] = v;
  }
}

// ---------------------------------------------------------------- driver

extern "C" void kernel_launch(void* const* d_in, const int* in_sizes, int n_in,
                              void* d_out, int out_size, void* d_ws, size_t ws_size,
                              hipStream_t stream) {
  const float*     x  = (const float*)d_in[0];
  const long long* ei = (const long long*)d_in[1];
  const float* W0c = (const float*)d_in[2];
  const float* b0c = (const float*)d_in[3];
  const float* W1c = (const float*)d_in[4];
  const float* b1c = (const float*)d_in[5];
  const float* W2c = (const float*)d_in[6];
  const float* b2c = (const float*)d_in[7];

  const int Fin = 64;
  int N = in_sizes[0] / Fin;        // 50000
  int E = in_sizes[1] / 2;          // 800000
  const long long* row = ei;
  const long long* col = ei + E;

  float* ws = (float*)d_ws;
  size_t off = 0;
  float* deg = ws + off; off += ((size_t)N + 15) & ~(size_t)15;
  float* w   = ws + off; off += ((size_t)E + 15) & ~(size_t)15;
  float* xd  = ws + off; off += (size_t)N * Fin;
  float* tx1 = ws + off; off += (size_t)N * Fin;
  float* h   = ws + off;            // layer hidden buffer (N x 64)

  const int T   = 256;
  int nel  = N * Fin;
  int nel4 = nel / 4;
  int n4   = N / 4;                 // 50000 % 4 == 0

  // edge weights: deg -> dis (in place) -> w
  zero4_f32 <<<(n4 + T - 1) / T, T, 0, stream>>>(deg, n4);
  deg_kernel<<<(E + T - 1) / T, T, 0, stream>>>(row, deg, E);
  dis_kernel<<<(N + T - 1) / T, T, 0, stream>>>(deg, N);
  wgt_kernel<<<(E + T - 1) / T, T, 0, stream>>>(row, col, deg, w, E);

  int pthreads = E * (Fin / 4);
  int rowTiles = N / 16;            // exact: 3125

  // ---- layer 0: x(64) -> h(64), ReLU
  dropout4_kernel<<<(nel4 + T - 1) / T, T, 0, stream>>>(x, xd, nel4, 0xC0FFEE00u);
  zero4_f32      <<<(nel4 + T - 1) / T, T, 0, stream>>>(tx1, nel4);
  prop_kernel    <<<(pthreads + T - 1) / T, T, 0, stream>>>(xd, row, col, w, tx1, E);
  cheb_gemm_wmma<64, 64, 1><<<rowTiles, 128, 0, stream>>>(
      xd, tx1, W0c, W0c + 64 * 64, b0c, h);

  // ---- layer 1: h(64) -> h(64), ReLU
  dropout4_kernel<<<(nel4 + T - 1) / T, T, 0, stream>>>(h, xd, nel4, 0xC0FFEE01u);
  zero4_f32      <<<(nel4 + T - 1) / T, T, 0, stream>>>(tx1, nel4);
  prop_kernel    <<<(pthreads + T - 1) / T, T, 0, stream>>>(xd, row, col, w, tx1, E);
  cheb_gemm_wmma<64, 64, 1><<<rowTiles, 128, 0, stream>>>(
      xd, tx1, W1c, W1c + 64 * 64, b1c, h);

  // ---- layer 2: h(64) -> out(32), no ReLU
  dropout4_kernel<<<(nel4 + T - 1) / T, T, 0, stream>>>(h, xd, nel4, 0xC0FFEE02u);
  zero4_f32      <<<(nel4 + T - 1) / T, T, 0, stream>>>(tx1, nel4);
  prop_kernel    <<<(pthreads + T - 1) / T, T, 0, stream>>>(xd, row, col, w, tx1, E);
  cheb_gemm_wmma<64, 32, 0><<<rowTiles, 64, 0, stream>>>(
      xd, tx1, W2c, W2c + 64 * 32, b2c, (float*)d_out);
}